// GeomLayer_41205916238178
// MI455X (gfx1250) — compile-verified
//
#include <hip/hip_runtime.h>

typedef __attribute__((ext_vector_type(2))) float v2f;
typedef __attribute__((ext_vector_type(8))) float v8f;
typedef __attribute__((ext_vector_type(4))) unsigned int v4u;
typedef __attribute__((ext_vector_type(4))) int v4i;
typedef __attribute__((ext_vector_type(8))) int v8i;

#define CHAN     256
#define IN_CHAN  512
#define BATCH    16
#define HH       56
#define WW       56
#define NPIX     (HH * WW)   // 3136

// ---------------------------------------------------------------------------
// LDS byte-offset of a __shared__ object (generic -> AS(3) addrspacecast).
// ---------------------------------------------------------------------------
static __device__ inline unsigned int lds_off(void* p) {
  return (unsigned int)(uintptr_t)(__attribute__((address_space(3))) void*)p;
}

// ---------------------------------------------------------------------------
// Issue a TDM 2D tile load: global (row-major, stride0 elements/row) -> LDS.
// Descriptor bit layout per CDNA5 ISA 8.3/8.4 (D# groups 0..3).
// data_size = 4 bytes. Optional LDS padding (interval/amount are encoded).
// ---------------------------------------------------------------------------
static __device__ inline void tdm_load_2d(unsigned int lds_byte,
                                          const void* gptr,
                                          unsigned int tensor_d0,
                                          unsigned int tensor_d1,
                                          unsigned int tile_d0,
                                          unsigned int tile_d1,
                                          unsigned long long stride0,
                                          int pad_en,
                                          unsigned int pad_interval_code,
                                          unsigned int pad_amount_code) {
  const unsigned long long ga = (unsigned long long)(uintptr_t)gptr;
  v4u g0;
  g0.x = 1u;                                             // count=1, no gather
  g0.y = lds_byte;                                       // lds_addr [63:32]
  g0.z = (unsigned int)ga;                               // global_addr lo
  g0.w = (unsigned int)((ga >> 32) & 0x1FFFFFFull)       // global_addr hi
         | (2u << 30);                                   // type=2 ("image")

  unsigned int d0 = (2u << 16);                          // data_size = 4B
  if (pad_en)
    d0 |= (1u << 20) | (pad_interval_code << 22) | (pad_amount_code << 25);
  v8i g1;
  g1.s0 = (int)d0;                                       // mask=0 (no cluster)
  g1.s1 = (int)((tensor_d0 & 0xFFFFu) << 16);            // barrier_addr=0
  g1.s2 = (int)((tensor_d0 >> 16) | ((tensor_d1 & 0xFFFFu) << 16));
  g1.s3 = (int)((tensor_d1 >> 16) | (tile_d0 << 16));
  g1.s4 = (int)tile_d1;                                  // tile_dim2 = 0 (2D)
  g1.s5 = (int)(unsigned int)(stride0 & 0xFFFFFFFFull);
  g1.s6 = (int)(unsigned int)((stride0 >> 32) & 0xFFFFull); // stride1 = 0
  g1.s7 = 0;

  v4i g2 = {0, 0, 0, 0};
  v4i g3 = {0, 0, 0, 0};
#if defined(__clang_major__) && (__clang_major__ >= 23)
  v8i g4 = {0, 0, 0, 0, 0, 0, 0, 0};
  __builtin_amdgcn_tensor_load_to_lds(g0, g1, g2, g3, g4, 0);
#else
  __builtin_amdgcn_tensor_load_to_lds(g0, g1, g2, g3, 0);
#endif
}

// ---------------------------------------------------------------------------
// Kernel 1: Y[b,c,p] = sum_i L[c,i] * X[b,i,p]    (per-batch 256x3136 GEMM)
// 256 threads = 8 waves; tile M=128, N=64, K-chunk = 32, TDM double-buffered.
// Each wave: 32x32 output = 2x2 tiles of v_wmma_f32_16x16x4_f32 accumulators.
// A staged with TDM pad (32 dwords + 1 pad dword) -> pitch-33, bank-clean.
// ---------------------------------------------------------------------------
__global__ __launch_bounds__(256)
void geom_gemm_wmma(const float* __restrict__ X,
                    const float* __restrict__ L,
                    float* __restrict__ Y) {
  __shared__ float As[2][128][33];   // TDM writes 32 data + 1 pad dword/row
  __shared__ float Bs[2][32][64];

  const int tid  = threadIdx.x;
  const int lane = tid & 31;
  const int wv   = tid >> 5;      // wave id 0..7
  const int wm   = wv >> 1;       // 0..3  (M sub-block of 32)
  const int wn   = wv & 1;        // 0..1  (N sub-block of 32)
  const int kh   = lane >> 4;     // half-wave selector (ISA A/B/D layouts)
  const int ml   = lane & 15;

  const int n_blk = blockIdx.x * 64;    // 49 N-blocks
  const int m_blk = blockIdx.y * 128;   // 2  M-blocks
  const int b     = blockIdx.z;         // 16 batches

  const float* __restrict__ Xb = X + (size_t)b * IN_CHAN * NPIX;

  unsigned int aOff[2] = { lds_off(&As[0][0][0]), lds_off(&As[1][0][0]) };
  unsigned int bOff[2] = { lds_off(&Bs[0][0][0]), lds_off(&Bs[1][0][0]) };

  // Prologue: wave 0 DMAs chunk 0 into buffer 0, drains TENSORcnt, barrier.
  if (tid < 32) {
    // A chunk: 128 rows x 32 k, row stride IN_CHAN; pad 1 dword per 32 dwords
    tdm_load_2d(aOff[0], &L[(size_t)m_blk * IN_CHAN],
                IN_CHAN, CHAN, 32u, 128u, (unsigned long long)IN_CHAN,
                1, 4u /* 32 dwords */, 0u /* 1 dword */);
    // B chunk: 32 k-rows x 64 cols, row stride NPIX, no pad
    tdm_load_2d(bOff[0], &Xb[n_blk],
                NPIX, IN_CHAN, 64u, 32u, (unsigned long long)NPIX,
                0, 0u, 0u);
    __builtin_amdgcn_s_wait_tensorcnt(0);
  }
  __syncthreads();

  v8f acc[2][2] = {};

  int buf = 0;
  for (int kc = 0; kc < IN_CHAN; kc += 32, buf ^= 1) {
    // Kick off next chunk's DMA into the other buffer (overlaps compute).
    if (tid < 32 && (kc + 32) < IN_CHAN) {
      const int kn = kc + 32;
      tdm_load_2d(aOff[buf ^ 1], &L[(size_t)m_blk * IN_CHAN + kn],
                  IN_CHAN, CHAN, 32u, 128u, (unsigned long long)IN_CHAN,
                  1, 4u, 0u);
      tdm_load_2d(bOff[buf ^ 1], &Xb[(size_t)kn * NPIX + n_blk],
                  NPIX, IN_CHAN, 64u, 32u, (unsigned long long)NPIX,
                  0, 0u, 0u);
    }

    #pragma unroll
    for (int ks = 0; ks < 32; ks += 4) {
      const int kk = ks + 2 * kh;
      v2f a0, a1, b0, b1;
      // A frag: lane(L) -> A[m = L&15][K = 2*(L>>4) + v]
      a0.x = As[buf][wm * 32 + ml][kk];          a0.y = As[buf][wm * 32 + ml][kk + 1];
      a1.x = As[buf][wm * 32 + 16 + ml][kk];     a1.y = As[buf][wm * 32 + 16 + ml][kk + 1];
      // B frag: lane(L) -> B[K = 2*(L>>4) + v][n = L&15]
      b0.x = Bs[buf][kk][wn * 32 + ml];          b0.y = Bs[buf][kk + 1][wn * 32 + ml];
      b1.x = Bs[buf][kk][wn * 32 + 16 + ml];     b1.y = Bs[buf][kk + 1][wn * 32 + 16 + ml];

      acc[0][0] = __builtin_amdgcn_wmma_f32_16x16x4_f32(false, a0, false, b0,
                    (short)0, acc[0][0], false, false);
      acc[0][1] = __builtin_amdgcn_wmma_f32_16x16x4_f32(false, a0, false, b1,
                    (short)0, acc[0][1], false, false);
      acc[1][0] = __builtin_amdgcn_wmma_f32_16x16x4_f32(false, a1, false, b0,
                    (short)0, acc[1][0], false, false);
      acc[1][1] = __builtin_amdgcn_wmma_f32_16x16x4_f32(false, a1, false, b1,
                    (short)0, acc[1][1], false, false);
    }

    // Drain this iteration's DMA, then release all waves into the next chunk.
    if (tid < 32) __builtin_amdgcn_s_wait_tensorcnt(0);
    __syncthreads();
  }

  // D layout: VGPR r holds row (r + 8*(L>>4)), col = L&15
  #pragma unroll
  for (int i = 0; i < 2; ++i)
    #pragma unroll
    for (int j = 0; j < 2; ++j)
      #pragma unroll
      for (int r = 0; r < 8; ++r) {
        const int row = m_blk + wm * 32 + i * 16 + r + 8 * kh;
        const int col = n_blk + wn * 32 + j * 16 + ml;
        Y[((size_t)b * CHAN + row) * NPIX + col] = acc[i][j][r];
      }
}

// ---------------------------------------------------------------------------
// Kernel 2: out[b,c,h,w] = bilinear(Y[b,c], theta_geo[c]) * box(theta_box[c])
// One block per (b,c) image. Full 56x56 image TDM-DMA'd into LDS (12.5 KB)
// while the waves set up per-channel affine parameters.
// box() is analytic (all-ones source) -> no second gather pass.
// ---------------------------------------------------------------------------
__global__ __launch_bounds__(256)
void geom_sample(const float* __restrict__ Y,
                 const float* __restrict__ geo,
                 const float* __restrict__ boxp,
                 float* __restrict__ Out) {
  __shared__ float img[NPIX];

  const int bc = blockIdx.x;          // b*CHAN + c
  const int c  = bc & (CHAN - 1);
  const float* __restrict__ src = Y + (size_t)bc * NPIX;

  // Single-row TDM descriptor: 3136 contiguous f32 -> LDS, overlapped with
  // the scalar theta loads below.
  if (threadIdx.x < 32) {
    tdm_load_2d(lds_off(&img[0]), src,
                NPIX, 1u, (unsigned int)NPIX, 1u, (unsigned long long)NPIX,
                0, 0u, 0u);
  }

  const float g00 = geo[c * 6 + 0], g01 = geo[c * 6 + 1], g02 = geo[c * 6 + 2];
  const float g10 = geo[c * 6 + 3], g11 = geo[c * 6 + 4], g12 = geo[c * 6 + 5];
  const float q00 = boxp[c * 6 + 0], q01 = boxp[c * 6 + 1], q02 = boxp[c * 6 + 2];
  const float q10 = boxp[c * 6 + 3], q11 = boxp[c * 6 + 4], q12 = boxp[c * 6 + 5];

  if (threadIdx.x < 32) __builtin_amdgcn_s_wait_tensorcnt(0);
  __syncthreads();

  for (int p = threadIdx.x; p < NPIX; p += 256) {
    const int h = p / WW, w = p % WW;
    const float xs = (2.f * (float)w + 1.f) / (float)WW - 1.f;
    const float ys = (2.f * (float)h + 1.f) / (float)HH - 1.f;

    // ---- geometric sample from LDS image ----
    const float gx = g00 * xs + g01 * ys + g02;
    const float gy = g10 * xs + g11 * ys + g12;
    const float ix = ((gx + 1.f) * (float)WW - 1.f) * 0.5f;
    const float iy = ((gy + 1.f) * (float)HH - 1.f) * 0.5f;
    const float fx0 = floorf(ix), fy0 = floorf(iy);
    const int x0 = (int)fx0, y0 = (int)fy0;
    const int x1 = x0 + 1,  y1 = y0 + 1;
    const float wx1 = ix - fx0, wx0 = 1.f - wx1;
    const float wy1 = iy - fy0, wy0 = 1.f - wy1;

    auto tap = [&](int yy, int xx) -> float {
      if (xx < 0 || xx >= WW || yy < 0 || yy >= HH) return 0.f;
      return img[yy * WW + xx];
    };
    const float samp = wy0 * (wx0 * tap(y0, x0) + wx1 * tap(y0, x1)) +
                       wy1 * (wx0 * tap(y1, x0) + wx1 * tap(y1, x1));

    // ---- analytic box factor (affine sample of an all-ones image) ----
    const float bx = q00 * xs + q01 * ys + q02;
    const float by = q10 * xs + q11 * ys + q12;
    const float jx = ((bx + 1.f) * (float)WW - 1.f) * 0.5f;
    const float jy = ((by + 1.f) * (float)HH - 1.f) * 0.5f;
    const float fbx0 = floorf(jx), fby0 = floorf(jy);
    const int bx0 = (int)fbx0, by0 = (int)fby0;
    const int bx1 = bx0 + 1,  by1 = by0 + 1;
    const float vx1 = jx - fbx0, vx0 = 1.f - vx1;
    const float vy1 = jy - fby0, vy0 = 1.f - vy1;

    auto ok = [&](int yy, int xx) -> float {
      return (xx >= 0 && xx < WW && yy >= 0 && yy < HH) ? 1.f : 0.f;
    };
    const float boxv = vy0 * (vx0 * ok(by0, bx0) + vx1 * ok(by0, bx1)) +
                       vy1 * (vx0 * ok(by1, bx0) + vx1 * ok(by1, bx1));

    Out[(size_t)bc * NPIX + p] = samp * boxv;
  }
}

// ---------------------------------------------------------------------------
extern "C" void kernel_launch(void* const* d_in, const int* in_sizes, int n_in,
                              void* d_out, int out_size, void* d_ws, size_t ws_size,
                              hipStream_t stream) {
  const float* x         = (const float*)d_in[0];  // [16,512,56,56]
  const float* linComb   = (const float*)d_in[1];  // [256,512]
  const float* geoParams = (const float*)d_in[2];  // [256,2,3]
  const float* boxParams = (const float*)d_in[3];  // [256,2,3]
  float* out = (float*)d_out;                      // [16,256,56,56]
  float* y   = (float*)d_ws;                       // scratch: 51.4 MB

  dim3 g1(NPIX / 64, CHAN / 128, BATCH);           // (49, 2, 16)
  geom_gemm_wmma<<<g1, 256, 0, stream>>>(x, linComb, y);

  geom_sample<<<dim3(BATCH * CHAN), 256, 0, stream>>>(y, geoParams, boxParams, out);
}